// NavierStokesPINN2_63221918597229
// MI455X (gfx1250) — compile-verified
//
#include <hip/hip_runtime.h>

// Navier-Stokes PINN forward + forward-mode derivatives for MI455X (gfx1250).
//
// Exploits exact piecewise-linearity of ReLU nets: second-order JVPs are
// identically zero (matches JAX's jvp-of-jvp through relu bit-for-bit), so we
// propagate 4 vectors per point (primal, d/dx, d/dy, d/dt) through the net.
// All hidden-layer GEMMs run on V_WMMA_F32_16X16X4_F32 (fp32 matrix pipe).

#define HDIM     512
#define LHID     6
#define PTS      32                   // points per workgroup
#define NVEC     4                    // primal, tx, ty, tt
#define MROWS    (PTS * NVEC)         // 128 GEMM rows per workgroup
#define MT       (MROWS / 16)         // 8 M-tiles
#define NWAVES   16
#define NTHREADS (NWAVES * 32)        // 512 threads
#define COLSW    (HDIM / NWAVES)      // 32 cols per wave
#define NT       (COLSW / 16)         // 2 N-tiles per wave

typedef __attribute__((ext_vector_type(2))) float v2f;
typedef __attribute__((ext_vector_type(8))) float v8f;

__global__ __launch_bounds__(NTHREADS)
void pinn_ns_kernel(const float* __restrict__ X, const float* __restrict__ Y,
                    const float* __restrict__ T,
                    const float* __restrict__ W_in, const float* __restrict__ b_in,
                    const float* __restrict__ W_hid, const float* __restrict__ b_hid,
                    const float* __restrict__ W_out, const float* __restrict__ b_out,
                    float* __restrict__ out, int N)
{
    extern __shared__ float smem[];
    float* act  = smem;                  // [MROWS][HDIM]  (256 KB)
    float* outv = smem + MROWS * HDIM;   // [MROWS][3]

    const int tid   = threadIdx.x;
    const int lane  = tid & 31;
    const int wave  = tid >> 5;
    const int lm    = lane & 15;          // M (A) / N (B,C,D) sub-index
    const int lk    = (lane >> 4) << 1;   // K offset: 0 (lanes 0-15) or 2 (16-31)
    const int hi8   = (lane >= 16) ? 8 : 0;
    const int ncol0 = wave * COLSW;
    const int pbase = blockIdx.x * PTS;

    // ---------------- layer 0: z = s @ W_in + b_in; tangents = W_in rows ----
    for (int idx = tid; idx < PTS * HDIM; idx += NTHREADS) {
        const int pt = idx >> 9;             // HDIM == 512
        const int u  = idx & (HDIM - 1);
        const int g  = pbase + pt;
        const float w0 = W_in[u], w1 = W_in[HDIM + u], w2 = W_in[2 * HDIM + u];
        const float zp = X[g] * w0 + Y[g] * w1 + T[g] * w2 + b_in[u];
        const bool  m  = zp > 0.0f;
        act[(0 * PTS + pt) * HDIM + u] = m ? zp : 0.0f;
        act[(1 * PTS + pt) * HDIM + u] = m ? w0 : 0.0f;
        act[(2 * PTS + pt) * HDIM + u] = m ? w1 : 0.0f;
        act[(3 * PTS + pt) * HDIM + u] = m ? w2 : 0.0f;
    }
    __syncthreads();

    // ---------------- hidden layers: [128x512] @ [512x512] via fp32 WMMA ----
    for (int layer = 0; layer < LHID; ++layer) {
        const float* __restrict__ Wl = W_hid + (size_t)layer * HDIM * HDIM;

        v8f acc[MT][NT] = {};

        for (int k = 0; k < HDIM; k += 4) {
            const int kb = k + lk;
            // A fragments (16x4 f32): lane = M, holds K=kb, kb+1   -> ds_load_b64
            v2f a[MT];
#pragma unroll
            for (int i = 0; i < MT; ++i)
                a[i] = *(const v2f*)&act[(i * 16 + lm) * HDIM + kb];
            // B fragments (4x16 f32): lane = N, holds K=kb, kb+1   -> global loads
            v2f b[NT];
#pragma unroll
            for (int j = 0; j < NT; ++j) {
                const int n = ncol0 + j * 16 + lm;
                b[j].x = Wl[(size_t)kb * HDIM + n];
                b[j].y = Wl[(size_t)(kb + 1) * HDIM + n];
            }
#pragma unroll
            for (int i = 0; i < MT; ++i)
#pragma unroll
                for (int j = 0; j < NT; ++j)
                    acc[i][j] = __builtin_amdgcn_wmma_f32_16x16x4_f32(
                        false, a[i], false, b[j], (short)0, acc[i][j],
                        false, false);
        }

        __syncthreads();   // all waves done reading act -> safe to overwrite
        // D layout: VGPR g -> (M = g + hi8, N = lm)
#pragma unroll
        for (int i = 0; i < MT; ++i)
#pragma unroll
            for (int j = 0; j < NT; ++j)
#pragma unroll
                for (int g = 0; g < 8; ++g) {
                    const int row = i * 16 + g + hi8;
                    act[row * HDIM + ncol0 + j * 16 + lm] = acc[i][j][g];
                }
        __syncthreads();

        // bias + ReLU; mask from primal gates all tangent rows
        const float* __restrict__ bl = b_hid + layer * HDIM;
        for (int idx = tid; idx < PTS * HDIM; idx += NTHREADS) {
            const int pt = idx >> 9;
            const int u  = idx & (HDIM - 1);
            const float zp = act[pt * HDIM + u] + bl[u];
            const bool  m  = zp > 0.0f;
            act[pt * HDIM + u] = m ? zp : 0.0f;
#pragma unroll
            for (int vv = 1; vv < NVEC; ++vv) {
                const int rr = (vv * PTS + pt) * HDIM + u;
                act[rr] = m ? act[rr] : 0.0f;
            }
        }
        __syncthreads();
    }

    // ---------------- output layer: 512 -> 3 (tiny; plain VALU dots) -------
    if (tid < MROWS * 3) {
        const int r = tid / 3, c = tid % 3;
        float s = (r < PTS) ? b_out[c] : 0.0f;   // bias only on primal rows
        for (int u = 0; u < HDIM; ++u)
            s += act[r * HDIM + u] * W_out[u * 3 + c];
        outv[tid] = s;
    }
    __syncthreads();

    // ---------------- NS residuals; u_xx = u_yy = v_xx = v_yy = 0 exactly --
    if (tid < PTS) {
        const int pt = tid, g = pbase + pt;
        const float u  = outv[pt * 3 + 0];
        const float v  = outv[pt * 3 + 1];
        const float p  = outv[pt * 3 + 2];  (void)p;
        const float ux = outv[(PTS + pt) * 3 + 0];
        const float vx = outv[(PTS + pt) * 3 + 1];
        const float px = outv[(PTS + pt) * 3 + 2];
        const float uy = outv[(2 * PTS + pt) * 3 + 0];
        const float vy = outv[(2 * PTS + pt) * 3 + 1];
        const float py = outv[(2 * PTS + pt) * 3 + 2];
        const float ut = outv[(3 * PTS + pt) * 3 + 0];
        const float vt = outv[(3 * PTS + pt) * 3 + 1];
        const float f  = ut + u * ux + v * uy + px;   // RHO=1; NU*(0+0)=0
        const float gg = vt + u * vx + v * vy + py;
        const float hh = ux + vy;
        out[0 * N + g] = u;
        out[1 * N + g] = v;
        out[2 * N + g] = outv[pt * 3 + 2];
        out[3 * N + g] = f;
        out[4 * N + g] = gg;
        out[5 * N + g] = hh;
    }
}

extern "C" void kernel_launch(void* const* d_in, const int* in_sizes, int n_in,
                              void* d_out, int out_size, void* d_ws, size_t ws_size,
                              hipStream_t stream) {
    (void)n_in; (void)out_size; (void)d_ws; (void)ws_size;
    const float* X     = (const float*)d_in[0];
    const float* Y     = (const float*)d_in[1];
    const float* T     = (const float*)d_in[2];
    const float* W_in  = (const float*)d_in[3];
    const float* b_in  = (const float*)d_in[4];
    const float* W_hid = (const float*)d_in[5];
    const float* b_hid = (const float*)d_in[6];
    const float* W_out = (const float*)d_in[7];
    const float* b_out = (const float*)d_in[8];
    float* out = (float*)d_out;
    const int N = in_sizes[0];

    const size_t smem = (size_t)(MROWS * HDIM + MROWS * 3) * sizeof(float); // ~258 KB
    hipFuncSetAttribute((const void*)pinn_ns_kernel,
                        hipFuncAttributeMaxDynamicSharedMemorySize, (int)smem);

    pinn_ns_kernel<<<N / PTS, NTHREADS, smem, stream>>>(
        X, Y, T, W_in, b_in, W_hid, b_hid, W_out, b_out, out, N);
}